// GraphConvModel_8392366096423
// MI455X (gfx1250) — compile-verified
//
#include <hip/hip_runtime.h>

typedef float v2f __attribute__((ext_vector_type(2)));
typedef float v8f __attribute__((ext_vector_type(8)));

#define D_IN  256
#define D_HID 64

// ---------------------------------------------------------------------------
// Degree accumulation: deg_src[src[e]] += 1 ; deg_dst[dst[e]] += 1
// ---------------------------------------------------------------------------
__global__ void deg_kernel(const int* __restrict__ src, const int* __restrict__ dst,
                           float* __restrict__ deg_src, float* __restrict__ deg_dst, int E) {
    int i = blockIdx.x * blockDim.x + threadIdx.x;
    if (i < E) {
        atomicAdd(&deg_src[src[i]], 1.0f);
        atomicAdd(&deg_dst[dst[i]], 1.0f);
    }
}

// In-place: d[i] = (max(d[i],1))^{-1/2}
__global__ void norm_kernel(float* __restrict__ d, int n) {
    int i = blockIdx.x * blockDim.x + threadIdx.x;
    if (i < n) d[i] = rsqrtf(fmaxf(d[i], 1.0f));
}

// ---------------------------------------------------------------------------
// Y[N x 64] = diag(rowscale) * X[N x K]  @  W[K x 64]   via V_WMMA_F32_16X16X4_F32
// Block = 128 threads = 4 waves (wave32). Block computes a 16-row stripe;
// wave w computes output columns [16w, 16w+16).
// A tile (16 x K, scaled) staged in LDS (row stride K+4 => stride % 64 == 4,
// so the 16 row-parallel lane reads hit distinct banks). W read from global
// (<= 64 KB, L2 / WGP$ resident across all 6250 blocks).
//
// Fragment layouts per CDNA5 ISA 7.12.2:
//   A 16x4 f32 : lane l<16 holds row l, K = k..k+1 ; lane l+16 holds row l, K = k+2..k+3
//   B 4x16 f32 : lane l<16 holds col l, K = k..k+1 ; lane l+16 holds col l, K = k+2..k+3
//   C/D 16x16  : VGPR r -> M=r (lanes 0-15) / M=8+r (lanes 16-31), N = lane&15
// ---------------------------------------------------------------------------
template <int K>
__global__ __launch_bounds__(128) void gemm_wmma_kernel(
        const float* __restrict__ X, const float* __restrict__ rowscale,
        const float* __restrict__ W, float* __restrict__ Y) {
    constexpr int SA = K + 4;               // padded LDS row stride for A
    __shared__ float lds_a[16 * SA];

    const int m0  = blockIdx.x * 16;
    const int tid = threadIdx.x;

    // Stage A tile (coalesced), folding in the per-row out-degree norm.
    for (int idx = tid; idx < 16 * K; idx += 128) {
        const int r = idx / K, c = idx - r * K;
        const float s = rowscale ? rowscale[m0 + r] : 1.0f;
        lds_a[r * SA + c] = X[(size_t)(m0 + r) * K + c] * s;
    }
    __syncthreads();

    const int wave = tid >> 5;
    const int lane = tid & 31;
    const int hi   = lane >> 4;             // 0: K+0..1, 1: K+2..3
    const int l    = lane & 15;             // row (A) / col (B) within tile
    const int n0   = wave * 16;

    v8f acc = {};
#pragma unroll 8
    for (int k = 0; k < K; k += 4) {
        const int koff = k + hi * 2;
        const v2f a = *(const v2f*)&lds_a[l * SA + koff];
        v2f b;
        b.x = W[(size_t)(koff    ) * D_HID + n0 + l];
        b.y = W[(size_t)(koff + 1) * D_HID + n0 + l];
        acc = __builtin_amdgcn_wmma_f32_16x16x4_f32(
                /*neg_a=*/false, a, /*neg_b=*/false, b,
                /*c_mod=*/(short)0, acc, /*reuse_a=*/false, /*reuse_b=*/false);
    }

    // D store: VGPR r -> row m0 + hi*8 + r, col n0 + l
    float* out = Y + (size_t)(m0 + hi * 8) * D_HID + n0 + l;
#pragma unroll
    for (int r = 0; r < 8; ++r) out[(size_t)r * D_HID] = acc[r];
}

// ---------------------------------------------------------------------------
// SpMM scatter: agg[dst[e], :] += H[src[e], :]   (64 feats, float4 per thread)
// 16 threads cooperate on one edge -> fully coalesced 256B gather of H[src].
// H and agg are 25.6 MB each: L2-resident, atomics never leave L2.
// ---------------------------------------------------------------------------
__global__ void scatter_kernel(const int* __restrict__ src, const int* __restrict__ dst,
                               const float* __restrict__ H, float* __restrict__ agg, int E) {
    long long t = (long long)blockIdx.x * blockDim.x + threadIdx.x;
    if (t >= (long long)E * 16) return;
    const int e = (int)(t >> 4);
    const int q = ((int)t & 15) * 4;
    const int s = src[e], d = dst[e];
    const float4 v = *(const float4*)&H[(size_t)s * D_HID + q];
    float* o = &agg[(size_t)d * D_HID + q];
    atomicAdd(o + 0, v.x);
    atomicAdd(o + 1, v.y);
    atomicAdd(o + 2, v.z);
    atomicAdd(o + 3, v.w);
}

// X2 = relu(agg * norm_dst + b1) * norm_src   (norm_src folded in for layer 2)
__global__ void finalize_relu_kernel(const float* __restrict__ agg,
                                     const float* __restrict__ norm_dst,
                                     const float* __restrict__ norm_src,
                                     const float* __restrict__ bias,
                                     float* __restrict__ X2, int n) {
    int t = blockIdx.x * blockDim.x + threadIdx.x;
    if (t < n * D_HID) {
        const int i = t >> 6, c = t & 63;
        const float v = agg[t] * norm_dst[i] + bias[c];
        X2[t] = fmaxf(v, 0.0f) * norm_src[i];
    }
}

// out = out * norm_dst + b2   (in place; out was zero-init'd then scatter-added)
__global__ void finalize_out_kernel(float* __restrict__ out,
                                    const float* __restrict__ norm_dst,
                                    const float* __restrict__ bias, int n) {
    int t = blockIdx.x * blockDim.x + threadIdx.x;
    if (t < n * D_HID) {
        const int i = t >> 6, c = t & 63;
        out[t] = out[t] * norm_dst[i] + bias[c];
    }
}

// ---------------------------------------------------------------------------
extern "C" void kernel_launch(void* const* d_in, const int* in_sizes, int n_in,
                              void* d_out, int out_size, void* d_ws, size_t ws_size,
                              hipStream_t stream) {
    const float* feat = (const float*)d_in[0];
    const int*   src  = (const int*)d_in[1];
    const int*   dst  = (const int*)d_in[2];
    const float* W1   = (const float*)d_in[3];
    const float* b1   = (const float*)d_in[4];
    const float* W2   = (const float*)d_in[5];
    const float* b2   = (const float*)d_in[6];
    float*       out  = (float*)d_out;

    const int N = in_sizes[0] / D_IN;   // 100000 (divisible by 16)
    const int E = in_sizes[1];          // 1.6M

    // Workspace layout (floats): [norm_src N][norm_dst N][bufA N*64][bufB N*64]
    float* ws       = (float*)d_ws;
    float* norm_src = ws;
    float* norm_dst = ws + N;
    float* bufA     = ws + 2 * (size_t)N;             // H1, then X2
    float* bufB     = bufA + (size_t)N * D_HID;       // agg1, then H2

    const int thr = 256;
    const long long scatter_work = (long long)E * 16;
    const int scatter_blocks = (int)((scatter_work + thr - 1) / thr);

    // Degrees -> norms
    hipMemsetAsync(norm_src, 0, sizeof(float) * 2 * (size_t)N, stream);
    deg_kernel<<<(E + thr - 1) / thr, thr, 0, stream>>>(src, dst, norm_src, norm_dst, E);
    norm_kernel<<<(2 * N + thr - 1) / thr, thr, 0, stream>>>(norm_src, 2 * N);

    // Layer 1: H1 = (feat * norm_src) @ W1
    gemm_wmma_kernel<D_IN><<<N / 16, 128, 0, stream>>>(feat, norm_src, W1, bufA);
    hipMemsetAsync(bufB, 0, sizeof(float) * (size_t)N * D_HID, stream);
    scatter_kernel<<<scatter_blocks, thr, 0, stream>>>(src, dst, bufA, bufB, E);
    // X2 = relu(agg * norm_dst + b1) * norm_src   (pre-scale for layer 2 GEMM)
    finalize_relu_kernel<<<(N * D_HID + thr - 1) / thr, thr, 0, stream>>>(
        bufB, norm_dst, norm_src, b1, bufA, N);

    // Layer 2: H2 = X2 @ W2 (norm already folded in -> rowscale = nullptr)
    gemm_wmma_kernel<D_HID><<<N / 16, 128, 0, stream>>>(bufA, nullptr, W2, bufB);
    hipMemsetAsync(out, 0, sizeof(float) * (size_t)N * D_HID, stream);
    scatter_kernel<<<scatter_blocks, thr, 0, stream>>>(src, dst, bufB, out, E);
    finalize_out_kernel<<<(N * D_HID + thr - 1) / thr, thr, 0, stream>>>(out, norm_dst, b2, N);
}